// CGNN_51333449121989
// MI455X (gfx1250) — compile-verified
//
#include <hip/hip_runtime.h>

#define NNODES 50000
#define EEDGES 600000
#define FIN    500
#define HDIM   64
#define SDIM   128   // 2*H ODE state width
#define CDIM   40
#define RK_STEPS 4

typedef __attribute__((ext_vector_type(2))) float        v2f;
typedef __attribute__((ext_vector_type(8))) float        v8f;
typedef __attribute__((ext_vector_type(4))) unsigned int v4u;
typedef __attribute__((ext_vector_type(8))) int          v8i;
typedef __attribute__((ext_vector_type(4))) int          v4i;

// ---------------------------------------------------------------------------
// Encoder: h = x @ m1_w + m1_b  (fp32 WMMA 16x16x4), then initialize the ODE
// state buffers: x0 = z = zs = acc = [h | 0], az = 0.
// Block = 128 threads (4 waves); block owns a 16-row tile; wave w owns output
// columns [16w, 16w+16). The 16x500 fp32 x-tile (32 KB) is DMA'd into LDS by
// the Tensor Data Mover (tensor_load_to_lds, TENSORcnt-tracked), issued by
// wave 0 only. The D# lds_addr is taken from &sx so the LDS array's address
// escapes into the side-effecting intrinsic — this keeps the ds_load reads
// of the tile alive in the optimizer.
// ---------------------------------------------------------------------------
__global__ __launch_bounds__(128)
void cgnn_encoder_wmma(const float* __restrict__ x,
                       const float* __restrict__ m1w,
                       const float* __restrict__ m1b,
                       float* __restrict__ x0,
                       float* __restrict__ z,
                       float* __restrict__ zs,
                       float* __restrict__ acc,
                       float* __restrict__ az)
{
    __shared__ float sx[16 * FIN];
    const int tid  = threadIdx.x;
    const int row0 = blockIdx.x * 16;

    // LDS byte offset of sx: low 32 bits of the generic pointer. This is the
    // correct D# field AND a ptrtoint escape of sx (prevents undef-folding of
    // the subsequent LDS loads).
    const unsigned lds_off = (unsigned)(uintptr_t)(void*)&sx[0];

    if (tid < 32) {                          // wave 0 issues the TDM descriptor
        // ---- D# group 0: count=1 | lds_addr | global_addr(57b) | type=2 ----
        unsigned long long ga =
            (unsigned long long)(uintptr_t)x + (unsigned long long)row0 * (FIN * 4ull);
        v4u g0;
        g0[0] = 1u;                          // count=1, is_restore=0, gather off
        g0[1] = lds_off;                     // lds_addr = offset of sx
        g0[2] = (unsigned int)ga;            // global_addr[31:0]
        g0[3] = (unsigned int)(ga >> 32) | (2u << 30);  // addr[56:32] | type=2
        // ---- D# group 1: data_size=4B, tensor 500 x 16, tile 500 x 16 ------
        v8i g1;
        g1[0] = (int)(2u << 16);             // data_size = 2 (4 bytes)
        g1[1] = (int)(500u << 16);           // tensor_dim0[15:0] = 500
        g1[2] = (int)(16u << 16);            // dim0 hi=0 | tensor_dim1[15:0]=16
        g1[3] = (int)(500u << 16);           // dim1 hi=0 | tile_dim0 = 500
        g1[4] = 16;                          // tile_dim1 = 16 (tile_dim2 = 0)
        g1[5] = 500;                         // tensor_dim0_stride[31:0] = 500
        g1[6] = 0;                           // stride hi | tensor_dim1_stride lo
        g1[7] = 0;
        v4i gz4 = {0, 0, 0, 0};              // groups 2/3 unused (2D tensor)
        v8i gz8 = {0, 0, 0, 0, 0, 0, 0, 0};  // trailing group (6-arg builtin)
        __builtin_amdgcn_tensor_load_to_lds(g0, g1, gz4, gz4, gz8, 0);
        __builtin_amdgcn_s_wait_tensorcnt(0);
        asm volatile("" ::: "memory");       // LDS now holds the tile
    }
    __syncthreads();                         // publish LDS tile to all 4 waves
    asm volatile("" ::: "memory");

    const int w    = tid >> 5;               // wave id -> N tile (H/16 = 4)
    const int l    = tid & 31;
    const int mr   = l & 15;                 // A row within tile
    const int koff = (l >> 4) << 1;          // 0 or 2 (K half per lane group)
    const int col  = (w << 4) + (l & 15);    // output column 0..63

    v8f c = {};
    for (int k0 = 0; k0 < FIN; k0 += 4) {
        v2f a, b;
        a.x = sx[mr * FIN + k0 + koff];
        a.y = sx[mr * FIN + k0 + koff + 1];
        b.x = m1w[(size_t)(k0 + koff)     * HDIM + col];
        b.y = m1w[(size_t)(k0 + koff + 1) * HDIM + col];
        c = __builtin_amdgcn_wmma_f32_16x16x4_f32(false, a, false, b,
                                                  (short)0, c, false, false);
    }

    const float bias = m1b[col];
    #pragma unroll
    for (int v = 0; v < 8; ++v) {
        int    m    = v + ((l >> 4) << 3);   // C/D row: v + 8*(lane>=16)
        size_t base = (size_t)(row0 + m) * SDIM;
        float  hv   = c[v] + bias;
        x0 [base + col] = hv;  x0 [base + HDIM + col] = 0.f;
        z  [base + col] = hv;  z  [base + HDIM + col] = 0.f;
        zs [base + col] = hv;  zs [base + HDIM + col] = 0.f;
        acc[base + col] = hv;  acc[base + HDIM + col] = 0.f;
        az [base + col] = 0.f; az [base + HDIM + col] = 0.f;
    }
}

// ---------------------------------------------------------------------------
// SpMM: az[src] += w * zs[dst], one wave32 per edge, one float4 per lane
// (32 lanes x 4 = 128-wide state). All state buffers are L2-resident on
// MI455X (128 MB working set < 192 MB L2): gathers and fp32 atomics on-chip.
// ---------------------------------------------------------------------------
__global__ __launch_bounds__(256)
void cgnn_spmm(const int* __restrict__ esrc, const int* __restrict__ edst,
               const float* __restrict__ ew,
               const float* __restrict__ zs, float* __restrict__ az, int ne)
{
    int e = blockIdx.x * 8 + (threadIdx.x >> 5);
    if (e >= ne) return;
    int   lane = threadIdx.x & 31;
    int   s = esrc[e], d = edst[e];
    float wgt = ew[e];
    const float4* zp = (const float4*)(zs + (size_t)d * SDIM);
    float4 v = zp[lane];
    float* ap = az + (size_t)s * SDIM + lane * 4;
    atomicAdd(ap + 0, wgt * v.x);
    atomicAdd(ap + 1, wgt * v.y);
    atomicAdd(ap + 2, wgt * v.z);
    atomicAdd(ap + 3, wgt * v.w);
}

// ---------------------------------------------------------------------------
// Fused RK4 stage (float4-vectorized): k = sigmoid(alpha)*0.5*(az - zs) + x0
//   acc += (wk*dt/6) * k                      (w6 precomputed on host)
//   zs   = z + cs*dt * k      (stages 0..2)   (csdt precomputed)
//   z=zs=acc                  (stage 3: commit step)
//   az   = 0  (ready for next SpMM — no separate memset launch)
// ---------------------------------------------------------------------------
__global__ __launch_bounds__(256)
void cgnn_stage(float4* __restrict__ az, const float4* __restrict__ x0,
                const float* __restrict__ alpha_train,
                float4* __restrict__ zs, float4* __restrict__ z,
                float4* __restrict__ acc,
                float w6, float csdt, int is_last, int total4)
{
    int i = blockIdx.x * 256 + threadIdx.x;
    if (i >= total4) return;
    int   node = i >> 5;                       // SDIM/4 = 32 float4 per node
    float alph = 1.0f / (1.0f + __expf(-alpha_train[node]));
    float4 azv = az[i], zsv = zs[i], x0v = x0[i], accv = acc[i];
    float4 k;
    k.x = alph * 0.5f * (azv.x - zsv.x) + x0v.x;
    k.y = alph * 0.5f * (azv.y - zsv.y) + x0v.y;
    k.z = alph * 0.5f * (azv.z - zsv.z) + x0v.z;
    k.w = alph * 0.5f * (azv.w - zsv.w) + x0v.w;
    accv.x += w6 * k.x;  accv.y += w6 * k.y;
    accv.z += w6 * k.z;  accv.w += w6 * k.w;
    acc[i] = accv;
    if (is_last) {
        z[i] = accv;  zs[i] = accv;
    } else {
        float4 zv = z[i];
        zs[i] = make_float4(zv.x + csdt * k.x, zv.y + csdt * k.y,
                            zv.z + csdt * k.z, zv.w + csdt * k.w);
    }
    az[i] = make_float4(0.f, 0.f, 0.f, 0.f);
}

// ---------------------------------------------------------------------------
// Decoder: out = relu(z[:, :H]) @ m2_w + m2_b  (fp32 WMMA, C=40 masked).
// Block = 96 threads (3 waves = 3 N-tiles covering 48 >= 40 columns).
// ---------------------------------------------------------------------------
__global__ __launch_bounds__(96)
void cgnn_decoder_wmma(const float* __restrict__ z,
                       const float* __restrict__ m2w,
                       const float* __restrict__ m2b,
                       float* __restrict__ out)
{
    const int tid  = threadIdx.x;
    const int row0 = blockIdx.x * 16;
    const int w    = tid >> 5;            // 0..2
    const int l    = tid & 31;
    const int mr   = l & 15;
    const int koff = (l >> 4) << 1;
    const int col  = (w << 4) + (l & 15); // 0..47
    const bool cok = (col < CDIM);

    v8f c = {};
    for (int k0 = 0; k0 < HDIM; k0 += 4) {
        v2f a, b;
        a.x = fmaxf(z[(size_t)(row0 + mr) * SDIM + k0 + koff],     0.f);
        a.y = fmaxf(z[(size_t)(row0 + mr) * SDIM + k0 + koff + 1], 0.f);
        b.x = cok ? m2w[(size_t)(k0 + koff)     * CDIM + col] : 0.f;
        b.y = cok ? m2w[(size_t)(k0 + koff + 1) * CDIM + col] : 0.f;
        c = __builtin_amdgcn_wmma_f32_16x16x4_f32(false, a, false, b,
                                                  (short)0, c, false, false);
    }
    if (cok) {
        float bias = m2b[col];
        #pragma unroll
        for (int v = 0; v < 8; ++v) {
            int m = v + ((l >> 4) << 3);
            out[(size_t)(row0 + m) * CDIM + col] = c[v] + bias;
        }
    }
}

// ---------------------------------------------------------------------------
extern "C" void kernel_launch(void* const* d_in, const int* in_sizes, int n_in,
                              void* d_out, int out_size, void* d_ws, size_t ws_size,
                              hipStream_t stream)
{
    const float* x    = (const float*)d_in[0];
    const float* ew   = (const float*)d_in[1];
    const float* m1w  = (const float*)d_in[2];
    const float* m1b  = (const float*)d_in[3];
    const float* alph = (const float*)d_in[4];
    const float* m2w  = (const float*)d_in[5];
    const float* m2b  = (const float*)d_in[6];
    const int*   esrc = (const int*)d_in[7];
    const int*   edst = (const int*)d_in[8];
    float* out = (float*)d_out;

    const size_t bufN = (size_t)NNODES * SDIM;   // floats per state buffer
    float* x0  = (float*)d_ws;
    float* z   = x0 + bufN;
    float* zs  = z  + bufN;
    float* az  = zs + bufN;
    float* acc = az + bufN;

    const float dt = 1.0f / RK_STEPS;

    cgnn_encoder_wmma<<<NNODES / 16, 128, 0, stream>>>(x, m1w, m1b,
                                                       x0, z, zs, acc, az);

    const int total4 = NNODES * SDIM / 4;
    const int gridS  = (total4 + 255) / 256;
    const int gridE  = (EEDGES + 7) / 8;
    const float wk[4] = {1.f, 2.f, 2.f, 1.f};
    const float cs[4] = {0.5f, 0.5f, 1.f, 0.f};

    for (int step = 0; step < RK_STEPS; ++step) {
        for (int s = 0; s < 4; ++s) {
            cgnn_spmm<<<gridE, 256, 0, stream>>>(esrc, edst, ew, zs, az, EEDGES);
            cgnn_stage<<<gridS, 256, 0, stream>>>((float4*)az, (const float4*)x0,
                                                  alph, (float4*)zs, (float4*)z,
                                                  (float4*)acc,
                                                  wk[s] * dt / 6.f, cs[s] * dt,
                                                  (s == 3) ? 1 : 0, total4);
        }
    }

    cgnn_decoder_wmma<<<NNODES / 16, 96, 0, stream>>>(z, m2w, m2b, out);
}